// PillarEncoder_30820685316296
// MI455X (gfx1250) — compile-verified
//
#include <hip/hip_runtime.h>
#include <hip/hip_bf16.h>
#include <math.h>

typedef __attribute__((ext_vector_type(2))) float v2f;
typedef __attribute__((ext_vector_type(8))) float v8f;

#define C_CH   64
#define M_PTS  32
#define H_IMG  192
#define W_IMG  640
#define XL     432
#define YL     496
#define VXC    0.16f
#define VYC    0.16f
#define X_OFF  0.08f
#define Y_OFF  (-39.6f)
#define BN_EPS 1e-3f
#define LN_EPS 1e-5f

// ---------------------------------------------------------------- zero canvas
__global__ void canvas_zero_kernel(float* __restrict__ out, long n) {
    long n4 = n >> 2;
    long i  = (long)blockIdx.x * blockDim.x + threadIdx.x;
    if (i < n4) ((float4*)out)[i] = make_float4(0.f, 0.f, 0.f, 0.f);
    if (i < (n & 3)) out[(n4 << 2) + i] = 0.f;
}

// --------------------------------------------- per-pillar: features+conv+pool
// one block of 64 threads per pillar; wave0 owns the 32 points
__global__ void __launch_bounds__(64)
pillar_kernel(const float* __restrict__ pillars,   // (P,32,4)
              const int*   __restrict__ coors,     // (P,3)
              const int*   __restrict__ npts,      // (P,)
              const float* __restrict__ image,     // (B,64,192,640)
              const float* __restrict__ calibs,    // (B,3,4)
              const float* __restrict__ conv_w,    // (64,9)
              const float* __restrict__ bn_gamma,
              const float* __restrict__ bn_beta,
              const float* __restrict__ bn_mean,
              const float* __restrict__ bn_var,
              float* __restrict__ pool_feat,       // (P,64)
              float* __restrict__ img_feat,        // (P,64)
              int P)
{
    __shared__ float feat[M_PTS][9];
    __shared__ float s_mean[3];
    __shared__ int   s_uvb[3];

    int p = blockIdx.x;
    if (p >= P) return;
    int t = threadIdx.x;

    int n   = npts[p];
    int cb0 = coors[p * 3 + 0];
    int cb1 = coors[p * 3 + 1];
    int cb2 = coors[p * 3 + 2];
    float cx = (float)cb1 * VXC + X_OFF;
    float cy = (float)cb2 * VYC + Y_OFF;

    float px = 0.f, py = 0.f, pz = 0.f, pr = 0.f;
    if (t < M_PTS) {
        const float* q = pillars + ((long)p * M_PTS + t) * 4;
        px = q[0]; py = q[1]; pz = q[2]; pr = q[3];
        // invalid points are already zero in the input -> sum is sum-of-valid
        float sx = px, sy = py, sz = pz;
        #pragma unroll
        for (int off = 16; off > 0; off >>= 1) {
            sx += __shfl_xor(sx, off, 32);
            sy += __shfl_xor(sy, off, 32);
            sz += __shfl_xor(sz, off, 32);
        }
        if (t == 0) {
            float inv = 1.f / (float)n;
            s_mean[0] = sx * inv; s_mean[1] = sy * inv; s_mean[2] = sz * inv;
        }
    }
    __syncthreads();

    float mx = s_mean[0], my = s_mean[1], mz = s_mean[2];
    if (t < M_PTS) {
        float valid = (t < n) ? 1.f : 0.f;
        float xo = px - cx, yo = py - cy;
        feat[t][0] = xo * valid;
        feat[t][1] = yo * valid;
        feat[t][2] = pz * valid;
        feat[t][3] = pr * valid;
        feat[t][4] = (px - mx) * valid;
        feat[t][5] = (py - my) * valid;
        feat[t][6] = (pz - mz) * valid;
        feat[t][7] = xo * valid;
        feat[t][8] = yo * valid;
    }
    __syncthreads();

    // channel t: 9-wide dot per point, BN (folded), ReLU, max over 32 points
    {
        float w[9];
        #pragma unroll
        for (int i = 0; i < 9; ++i) w[i] = conv_w[t * 9 + i];
        float scale = bn_gamma[t] * rsqrtf(bn_var[t] + BN_EPS);
        float shift = bn_beta[t] - bn_mean[t] * scale;
        float mv = 0.f;                   // relu outputs are >= 0
        #pragma unroll 4
        for (int m = 0; m < M_PTS; ++m) {
            float h = 0.f;
            #pragma unroll
            for (int i = 0; i < 9; ++i) h = fmaf(feat[m][i], w[i], h);
            h  = fmaf(h, scale, shift);
            mv = fmaxf(mv, fmaxf(h, 0.f));
        }
        pool_feat[(long)p * C_CH + t] = mv;
    }

    // projection (thread 0) runs concurrently with the conv above
    if (t == 0) {
        const float* K = calibs + (long)cb0 * 12;
        float p0 = K[0] * mx + K[1] * my + K[2]  * mz + K[3];
        float p1 = K[4] * mx + K[5] * my + K[6]  * mz + K[7];
        float p2 = K[8] * mx + K[9] * my + K[10] * mz + K[11];
        float inv_d = 1.f / p2;
        float u = fminf(fmaxf(p0 * inv_d * 0.5f, 0.f), (float)(W_IMG - 1));
        float v = fminf(fmaxf(p1 * inv_d * 0.5f, 0.f), (float)(H_IMG - 1));
        s_uvb[0] = (int)u;
        s_uvb[1] = (int)v;
        s_uvb[2] = cb0;
    }
    __syncthreads();

    {
        int u = s_uvb[0], v = s_uvb[1], b = s_uvb[2];
        long idx = (((long)b * C_CH + t) * H_IMG + v) * W_IMG + u;
        img_feat[(long)p * C_CH + t] = image[idx];
    }
}

// ------------------------------------------------- gating MLP via f32 WMMA
// one wave (32 lanes) per 16 pillars
__global__ void __launch_bounds__(32)
mlp_kernel(const float* __restrict__ pool_feat,  // (P,64)
           const float* __restrict__ img_feat,   // (P,64)
           const int*   __restrict__ coors,      // (P,3)
           const float* __restrict__ g_w1,       // (128,128)
           const float* __restrict__ g_b1,       // (128,)
           const float* __restrict__ ln_g,       // (128,)
           const float* __restrict__ ln_b,       // (128,)
           const float* __restrict__ g_w2,       // (64,128)
           const float* __restrict__ g_b2,       // (64,)
           float* __restrict__ out,              // (B,64,YL,XL)
           int P)
{
    __shared__ float Xs[16][132];   // padded: stride 132 -> conflict-free rows
    __shared__ float Gs[16][132];
    __shared__ int   cb[16][3];

    int lane = threadIdx.x;
    int p0   = blockIdx.x * 16;
    int half = lane >> 4;
    int l16  = lane & 15;

    // stage X = [pool | img] (16 x 128) into LDS (clamp tail rows)
    for (int r = 0; r < 16; ++r) {
        int  pp   = min(p0 + r, P - 1);
        long base = (long)pp * C_CH;
        for (int c = lane; c < C_CH; c += 32) {
            Xs[r][c]          = pool_feat[base + c];
            Xs[r][C_CH + c]   = img_feat[base + c];
        }
    }
    if (lane < 16) {
        int pp = min(p0 + lane, P - 1);
        cb[lane][0] = coors[pp * 3 + 0];
        cb[lane][1] = coors[pp * 3 + 1];
        cb[lane][2] = coors[pp * 3 + 2];
    }
    __syncthreads();

    // ---- layer 1: G1 = X @ W1^T + b1   (16x128 @ 128x128)
    for (int nt = 0; nt < 8; ++nt) {
        float bias = g_b1[nt * 16 + l16];
        v8f acc;
        #pragma unroll
        for (int i = 0; i < 8; ++i) acc[i] = bias;
        const float* wrow = g_w1 + (long)(nt * 16 + l16) * 128;  // B col = lane
        #pragma unroll
        for (int kt = 0; kt < 32; ++kt) {
            int k = kt * 4 + half * 2;
            v2f a, b;
            a.x = Xs[l16][k];   a.y = Xs[l16][k + 1];
            b.x = wrow[k];      b.y = wrow[k + 1];
            acc = __builtin_amdgcn_wmma_f32_16x16x4_f32(
                      false, a, false, b, (short)0, acc, false, false);
        }
        #pragma unroll
        for (int v = 0; v < 8; ++v)
            Gs[v + 8 * half][nt * 16 + l16] = acc[v];
    }
    __syncthreads();

    // ---- layernorm + relu per row (lane pair (l16, l16+16) shares a row)
    {
        float s = 0.f, s2 = 0.f;
        #pragma unroll 8
        for (int j = half * 64; j < half * 64 + 64; ++j) {
            float g = Gs[l16][j];
            s += g; s2 += g * g;
        }
        s  += __shfl_xor(s, 16, 32);
        s2 += __shfl_xor(s2, 16, 32);
        float mu   = s * (1.f / 128.f);
        float var  = s2 * (1.f / 128.f) - mu * mu;
        float rstd = rsqrtf(var + LN_EPS);
        #pragma unroll 8
        for (int j = half * 64; j < half * 64 + 64; ++j) {
            float g = (Gs[l16][j] - mu) * rstd * ln_g[j] + ln_b[j];
            Gs[l16][j] = fmaxf(g, 0.f);
        }
    }
    __syncthreads();

    // ---- layer 2: logits = G @ W2^T + b2 ; sigmoid gate ; blend ; scatter
    for (int nt = 0; nt < 4; ++nt) {
        float bias = g_b2[nt * 16 + l16];
        v8f acc;
        #pragma unroll
        for (int i = 0; i < 8; ++i) acc[i] = bias;
        const float* wrow = g_w2 + (long)(nt * 16 + l16) * 128;
        #pragma unroll
        for (int kt = 0; kt < 32; ++kt) {
            int k = kt * 4 + half * 2;
            v2f a, b;
            a.x = Gs[l16][k];   a.y = Gs[l16][k + 1];
            b.x = wrow[k];      b.y = wrow[k + 1];
            acc = __builtin_amdgcn_wmma_f32_16x16x4_f32(
                      false, a, false, b, (short)0, acc, false, false);
        }
        int c = nt * 16 + l16;
        #pragma unroll
        for (int v = 0; v < 8; ++v) {
            int r = v + 8 * half;
            if (p0 + r >= P) continue;
            float gate = 1.f / (1.f + __expf(-acc[v]));
            float val  = Xs[r][c] * gate + Xs[r][C_CH + c] * (1.f - gate);
            int bb = cb[r][0], x = cb[r][1], y = cb[r][2];
            out[(((long)bb * C_CH + c) * YL + y) * XL + x] = val;
        }
    }
}

// ----------------------------------------------------------------- launcher
extern "C" void kernel_launch(void* const* d_in, const int* in_sizes, int n_in,
                              void* d_out, int out_size, void* d_ws, size_t ws_size,
                              hipStream_t stream) {
    (void)n_in; (void)ws_size;
    const float* pillars  = (const float*)d_in[0];
    const int*   coors    = (const int*)  d_in[1];
    const int*   npts     = (const int*)  d_in[2];
    const float* image    = (const float*)d_in[3];
    const float* calibs   = (const float*)d_in[4];
    /* d_in[5] = batch_size (device scalar, unused on host) */
    const float* conv_w   = (const float*)d_in[6];
    const float* bn_gamma = (const float*)d_in[7];
    const float* bn_beta  = (const float*)d_in[8];
    const float* bn_mean  = (const float*)d_in[9];
    const float* bn_var   = (const float*)d_in[10];
    const float* g_w1     = (const float*)d_in[11];
    const float* g_b1     = (const float*)d_in[12];
    const float* ln_g     = (const float*)d_in[13];
    const float* ln_b     = (const float*)d_in[14];
    const float* g_w2     = (const float*)d_in[15];
    const float* g_b2     = (const float*)d_in[16];

    int   P   = in_sizes[0] / (M_PTS * 4);
    float* out       = (float*)d_out;
    float* pool_feat = (float*)d_ws;
    float* img_feat  = pool_feat + (size_t)P * C_CH;

    long n  = (long)out_size;
    long n4 = n >> 2;
    int  zb = (int)((n4 + 255) / 256);
    canvas_zero_kernel<<<zb, 256, 0, stream>>>(out, n);

    pillar_kernel<<<P, 64, 0, stream>>>(pillars, coors, npts, image, calibs,
                                        conv_w, bn_gamma, bn_beta, bn_mean,
                                        bn_var, pool_feat, img_feat, P);

    int mb = (P + 15) / 16;
    mlp_kernel<<<mb, 32, 0, stream>>>(pool_feat, img_feat, coors,
                                      g_w1, g_b1, ln_g, ln_b, g_w2, g_b2,
                                      out, P);
}